// CausalAttentionLayer_78700980731997
// MI455X (gfx1250) — compile-verified
//
#include <hip/hip_runtime.h>
#include <hip/hip_bf16.h>
#include <math.h>

// CDNA5 wave32 fp32 WMMA attention for MI455X.
//   N_C = N_E = 8192, IN_FEAT = 256, D = OUT_FEAT = 128.
// Outputs (concat in d_out): causal_output [8192*128] f32, attn_weights [8192*8192] f32.
//
// Bandwidth plan (~1 GB @ 23.3 TB/s => ~45us floor):
//   mask read 256MB + unnormalized attn write 256MB + attn re-read/re-write 512MB.
// Column dimension is split into up to 8 segments so the streaming kernel runs
// 4096 waves (vs 512), enough to keep HBM saturated. Per-segment partial row
// sums / partial O go to workspace and are reduced deterministically.

#define N_C       8192
#define N_E       8192
#define IN_FEAT   256
#define OUT_FEAT  128

typedef __attribute__((ext_vector_type(2))) float v2f;
typedef __attribute__((ext_vector_type(8))) float v8f;

// D(16x16,f32) = A(16x4,f32) x B(4x16,f32) + C   (V_WMMA_F32_16X16X4_F32)
// A layout: lane = m + 16*(k>=2), vgpr{0,1} = k pair  -> each lane: float2 at [m][2*hi]
// B layout: lane = n + 16*(k>=2), vgpr{0,1} = k pair  -> each lane: B[2*hi][n], B[2*hi+1][n]
// C layout: lane = n + 16*(m>=8), vgpr r = row (r + 8*hi)
static __device__ __forceinline__ v8f wmma_f32(v2f a, v2f b, v8f c) {
    return __builtin_amdgcn_wmma_f32_16x16x4_f32(false, a, false, b, (short)0, c, false, false);
}

// ---------------------------------------------------------------------------
// Kernel 1: QKV projection.  out = emb @ W + b   ([8192,256] x [256,128])
// One wave per 16x16 output tile; 64 WMMA steps over K=256.
// ---------------------------------------------------------------------------
__global__ void qkv_kernel(const float* __restrict__ emb_cause,
                           const float* __restrict__ emb_effect,
                           const float* __restrict__ Wq, const float* __restrict__ bq,
                           const float* __restrict__ Wk, const float* __restrict__ bk,
                           const float* __restrict__ Wv, const float* __restrict__ bv,
                           float* __restrict__ Qo, float* __restrict__ Ko,
                           float* __restrict__ Vo)
{
    const float* emb; const float* W; const float* bias; float* out;
    if (blockIdx.y == 0)      { emb = emb_cause;  W = Wq; bias = bq; out = Qo; }
    else if (blockIdx.y == 1) { emb = emb_effect; W = Wk; bias = bk; out = Ko; }
    else                      { emb = emb_effect; W = Wv; bias = bv; out = Vo; }

    const int wave = threadIdx.x >> 5;
    const int lane = threadIdx.x & 31;
    const int ln   = lane & 15;      // m for A, n for B/C
    const int hi   = lane >> 4;

    const int tile = blockIdx.x * 4 + wave;          // 4096 tiles per matrix
    const int tm   = tile >> 3;                      // 512 row tiles
    const int tn   = tile & 7;                       // 8 col tiles

    const float* arow = emb + (size_t)(tm * 16 + ln) * IN_FEAT + 2 * hi;
    const float* bcol = W + tn * 16 + ln;            // + k*OUT_FEAT per k

    v8f c = {};
#pragma unroll
    for (int k0 = 0; k0 < IN_FEAT; k0 += 4) {
        v2f a = *(const v2f*)(arow + k0);
        v2f b;
        b.x = bcol[(k0 + 2 * hi)     * OUT_FEAT];
        b.y = bcol[(k0 + 2 * hi + 1) * OUT_FEAT];
        c = wmma_f32(a, b, c);
    }

    const int col = tn * 16 + ln;
    const float bb = bias[col];
#pragma unroll
    for (int r = 0; r < 8; ++r) {
        const int row = tm * 16 + r + 8 * hi;
        out[(size_t)row * OUT_FEAT + col] = c[r] + bb;
    }
}

// ---------------------------------------------------------------------------
// Kernel 2: fused scores + masked exp + unnormalized attn write + partial
// row-sums and partial O = E @ V, per (16-row tile, column segment).
// One wave = 16 rows x one segment of columns.
// ---------------------------------------------------------------------------
__global__ void attn_part_kernel(const float* __restrict__ Q,
                                 const float* __restrict__ K,
                                 const float* __restrict__ V,
                                 const int*   __restrict__ mask,
                                 float* __restrict__ attn,       // [N_C, N_E] unnormalized e
                                 float* __restrict__ O_part,     // [nseg, N_C, OUT_FEAT]
                                 float* __restrict__ lsum_part,  // [nseg, N_C]
                                 int segLen)
{
    __shared__ float ldsE[4][16][16];   // per-wave 16x16 E tile for C->A layout swap

    const int wave = threadIdx.x >> 5;
    const int lane = threadIdx.x & 31;
    const int ln   = lane & 15;
    const int hi   = lane >> 4;

    const int seg      = blockIdx.y;
    const int colStart = seg * segLen;
    const int rowBase  = (blockIdx.x * 4 + wave) * 16;
    const float scale  = 0.088388347648318447f;   // 1/sqrt(128)

    // Preload this wave's 16x128 Q stripe into registers (A-layout fragments).
    v2f q[OUT_FEAT / 4];
    {
        const float* qrow = Q + (size_t)(rowBase + ln) * OUT_FEAT + 2 * hi;
#pragma unroll
        for (int i = 0; i < OUT_FEAT / 4; ++i) q[i] = *(const v2f*)(qrow + 4 * i);
    }

    v8f O[8];
    v8f zero = {};
#pragma unroll
    for (int nc = 0; nc < 8; ++nc) O[nc] = zero;
    float lsum[8] = {0.f, 0.f, 0.f, 0.f, 0.f, 0.f, 0.f, 0.f};

    const int nchunks = segLen >> 4;
    for (int j = 0; j < nchunks; ++j) {
        const int colBase = colStart + j * 16;

        // ---- S tile: 16x16 = Q(16x128) . K(16x128)^T via 32 f32 WMMAs ----
        v8f c = zero;
        const float* krow = K + (size_t)(colBase + ln) * OUT_FEAT + 2 * hi;
#pragma unroll
        for (int k0 = 0; k0 < OUT_FEAT; k0 += 4) {
            v2f b = *(const v2f*)(krow + k0);       // B = K^T: same fragment pattern as A
            c = wmma_f32(q[k0 >> 2], b, c);
        }

        // ---- mask + exp, write unnormalized weights, row sums, stage E ----
#pragma unroll
        for (int r = 0; r < 8; ++r) {
            const int row = rowBase + r + 8 * hi;
            const int col = colBase + ln;
            float e = 0.0f;
            if (mask[(size_t)row * N_E + col] != 0) e = __expf(c[r] * scale);
            lsum[r] += e;
            attn[(size_t)row * N_E + col] = e;
            ldsE[wave][r + 8 * hi][ln] = e;         // E[mrow][ncol]
        }

        // ---- O(16x128) += E(16x16) . V[colBase:colBase+16, :] ----
#pragma unroll
        for (int nc = 0; nc < 8; ++nc) {
            v8f acc = O[nc];
#pragma unroll
            for (int ks = 0; ks < 4; ++ks) {
                v2f a = *(const v2f*)(&ldsE[wave][ln][4 * ks + 2 * hi]);  // A-layout read
                const int kk = colBase + 4 * ks + 2 * hi;
                v2f b;
                b.x = V[(size_t)kk       * OUT_FEAT + nc * 16 + ln];
                b.y = V[(size_t)(kk + 1) * OUT_FEAT + nc * 16 + ln];
                acc = wmma_f32(a, b, acc);
            }
            O[nc] = acc;
        }
    }

    // ---- reduce row sums across the 16 lanes sharing a row (stays in half) ----
    float lred[8];
#pragma unroll
    for (int r = 0; r < 8; ++r) {
        float s = lsum[r];
        s += __shfl_xor(s, 1, 32);
        s += __shfl_xor(s, 2, 32);
        s += __shfl_xor(s, 4, 32);
        s += __shfl_xor(s, 8, 32);
        lred[r] = s;
    }

    if (ln == 0) {
#pragma unroll
        for (int r = 0; r < 8; ++r)
            lsum_part[(size_t)seg * N_C + rowBase + r + 8 * hi] = lred[r];
    }

    // ---- partial (unnormalized) O write for this segment ----
#pragma unroll
    for (int nc = 0; nc < 8; ++nc) {
#pragma unroll
        for (int r = 0; r < 8; ++r) {
            const int row = rowBase + r + 8 * hi;
            O_part[((size_t)seg * N_C + row) * OUT_FEAT + nc * 16 + ln] = O[nc][r];
        }
    }
}

// ---------------------------------------------------------------------------
// Kernel 3: deterministic reduction of segment partials.
//   rowsum = sum_s lsum_part[s][row];  rowinv = 1/rowsum
//   out_causal[row][col] = (sum_s O_part[s][row][col]) * rowinv
// ---------------------------------------------------------------------------
__global__ void finalize_kernel(const float* __restrict__ O_part,
                                const float* __restrict__ lsum_part,
                                float* __restrict__ out_causal,
                                float* __restrict__ rowinv,
                                int nseg)
{
    const int idx = blockIdx.x * blockDim.x + threadIdx.x;   // < N_C*OUT_FEAT
    const int row = idx >> 7;
    const int col = idx & (OUT_FEAT - 1);

    float rs = 0.0f;
    for (int s = 0; s < nseg; ++s) rs += lsum_part[(size_t)s * N_C + row];
    const float rinv = 1.0f / rs;

    float acc = 0.0f;
    for (int s = 0; s < nseg; ++s)
        acc += O_part[((size_t)s * N_C + row) * OUT_FEAT + col];

    out_causal[idx] = acc * rinv;
    if (col == 0) rowinv[row] = rinv;
}

// ---------------------------------------------------------------------------
// Kernel 4: attn[row, :] *= rowinv[row]   (float4 streaming, row len % 4 == 0)
// ---------------------------------------------------------------------------
__global__ void norm_kernel(float* __restrict__ attn, const float* __restrict__ rowinv)
{
    const size_t idx = ((size_t)blockIdx.x * blockDim.x + threadIdx.x) * 4;
    if (idx < (size_t)N_C * N_E) {
        float4 v = *(float4*)(attn + idx);
        const float s = rowinv[idx >> 13];   // idx / 8192
        v.x *= s; v.y *= s; v.z *= s; v.w *= s;
        *(float4*)(attn + idx) = v;
    }
}

// ---------------------------------------------------------------------------
extern "C" void kernel_launch(void* const* d_in, const int* in_sizes, int n_in,
                              void* d_out, int out_size, void* d_ws, size_t ws_size,
                              hipStream_t stream) {
    const float* emb_cause   = (const float*)d_in[0];
    const float* emb_effect  = (const float*)d_in[1];
    const int*   causal_mask = (const int*)d_in[2];
    const float* Wq = (const float*)d_in[3];
    const float* bq = (const float*)d_in[4];
    const float* Wk = (const float*)d_in[5];
    const float* bk = (const float*)d_in[6];
    const float* Wv = (const float*)d_in[7];
    const float* bv = (const float*)d_in[8];

    float* out_causal = (float*)d_out;                                   // [8192,128]
    float* out_attn   = (float*)d_out + (size_t)N_C * OUT_FEAT;          // [8192,8192]

    const size_t mat = (size_t)N_C * OUT_FEAT;                           // 1M floats

    // Pick the largest column-segment count whose partial buffers fit in ws.
    const size_t base_f    = 3 * mat + N_C;          // Q, K, V, rowinv
    const size_t per_seg_f = mat + N_C;              // O_part + lsum_part per segment
    int nseg = 1;
    for (int s = 8; s >= 1; s >>= 1) {
        if ((base_f + (size_t)s * per_seg_f) * sizeof(float) <= ws_size) { nseg = s; break; }
    }

    float* Q         = (float*)d_ws;
    float* K         = Q + mat;
    float* V         = K + mat;
    float* rowinv    = V + mat;
    float* lsum_part = rowinv + N_C;
    float* O_part    = lsum_part + (size_t)nseg * N_C;

    // QKV projection: 4096 16x16 tiles per matrix, 4 waves/block, y = {Q,K,V}
    qkv_kernel<<<dim3(1024, 3), 128, 0, stream>>>(
        emb_cause, emb_effect, Wq, bq, Wk, bk, Wv, bv, Q, K, V);

    // Fused attention partials: 128 row-groups x nseg segments, 4 waves/block.
    const int segLen = N_E / nseg;
    attn_part_kernel<<<dim3(128, nseg), 128, 0, stream>>>(
        Q, K, V, causal_mask, out_attn, O_part, lsum_part, segLen);

    // Reduce partials -> causal_output and rowinv.  1M elements / 256.
    finalize_kernel<<<dim3((int)(mat / 256)), 256, 0, stream>>>(
        O_part, lsum_part, out_causal, rowinv, nseg);

    // Normalize attention weights: 64M elements / 4 per thread / 256 per block.
    norm_kernel<<<dim3(65536), 256, 0, stream>>>(out_attn, rowinv);
}